// SAINT_87488483820171
// MI455X (gfx1250) — compile-verified
//
#include <hip/hip_runtime.h>
#include <hip/hip_bf16.h>

#define N_NODES 100000
#define N_EDGES 1600000
#define HIDC 64
#define OUTC 40

typedef __attribute__((ext_vector_type(2))) float v2f;
typedef __attribute__((ext_vector_type(8))) float v8f;

// ---------------------------------------------------------------- utilities
__global__ void zero_f32(float* __restrict__ p, int n) {
    int i = blockIdx.x * blockDim.x + threadIdx.x;
    if (i < n) p[i] = 0.0f;
}

__global__ void degree_kernel(const int* __restrict__ eidx, float* __restrict__ deg, int nEdges) {
    int e = blockIdx.x * blockDim.x + threadIdx.x;
    if (e < nEdges) atomicAdd(&deg[eidx[e]], 1.0f);   // rows = eidx[0..E)
}

__global__ void rsqrt_kernel(float* __restrict__ deg, int n) {
    int i = blockIdx.x * blockDim.x + threadIdx.x;
    if (i < n) {
        float d = deg[i];
        deg[i] = (d > 0.0f) ? rsqrtf(d) : 0.0f;       // d^-1/2, 0 for isolated nodes
    }
}

__global__ void relu_kernel(float* __restrict__ x, int n) {
    int i = blockIdx.x * blockDim.x + threadIdx.x;
    if (i < n) x[i] = fmaxf(x[i], 0.0f);
}

// ------------------------------------------------- WMMA fp32 GEMM: OUT = X @ W^T + bias
// X: [M x 64] row-major, W: [64 x 64] row-major ([out][in]), OUT: [M x 64].
// One wave per 16x16 output tile; K=64 -> 16x V_WMMA_F32_16X16X4_F32.
// A frag (16x4 f32, 2 VGPR): lanes 0-15 hold row M=lane, K = k0+0/1; lanes 16-31 hold K = k0+2/3.
// B frag (4x16 f32, 2 VGPR): lanes 0-15 hold col N=lane, K = k0+0/1; lanes 16-31 hold K = k0+2/3.
// C/D (16x16 f32, 8 VGPR): VGPR v = row M = v + 8*(lane>=16), col N = lane&15.
__global__ __launch_bounds__(256)
void gemm64_wmma(const float* __restrict__ X, const float* __restrict__ W,
                 const float* __restrict__ bias, float* __restrict__ out) {
    int lane = threadIdx.x & 31;
    int wave = threadIdx.x >> 5;
    int tile = blockIdx.x * 8 + wave;     // 25000 tiles total, 8 waves/block, exact
    int ct   = tile & 3;                  // 4 column tiles of 16 (64 outputs)
    int rt   = tile >> 2;                 // 6250 row tiles (100000/16 exact)
    int m0   = rt * 16;
    int n0   = ct * 16;
    int half = lane >> 4;                 // 0: K pair {0,1}; 1: K pair {2,3}
    int l15  = lane & 15;

    const float* Arow = X + (size_t)(m0 + l15) * HIDC + half * 2;
    const float* Brow = W + (size_t)(n0 + l15) * HIDC + half * 2;   // B[k][n] = W[n][k]

    v8f acc = {};
#pragma unroll
    for (int k0 = 0; k0 < HIDC; k0 += 4) {
        v2f a = *(const v2f*)(Arow + k0);
        v2f b = *(const v2f*)(Brow + k0);
        acc = __builtin_amdgcn_wmma_f32_16x16x4_f32(false, a, false, b,
                                                    (short)0, acc, false, false);
    }

    int   col   = n0 + l15;
    float bv    = bias[col];
    int   rbase = m0 + half * 8;
#pragma unroll
    for (int v = 0; v < 8; ++v) {
        out[(size_t)(rbase + v) * HIDC + col] = acc[v] + bv;
    }
}

// ------------------------------------------------- edge scatter: out[row] += w * h[col]
// 16 threads per edge; each thread handles 4 consecutive features (float4 gather + 4 f32 atomics).
__global__ __launch_bounds__(256)
void scatter_kernel(const int* __restrict__ eidx, const float* __restrict__ dis,
                    const float* __restrict__ h, float* __restrict__ out) {
    unsigned gid = blockIdx.x * 256u + threadIdx.x;
    int e  = (int)(gid >> 4);
    int fo = ((int)gid & 15) * 4;
    if (e >= N_EDGES) return;
    int r = eidx[e];
    int c = eidx[N_EDGES + e];
    float w = dis[r] * dis[c];
    const float4 hv = *(const float4*)(h + (size_t)c * HIDC + fo);
    float* o = out + (size_t)r * HIDC + fo;
    atomicAdd(o + 0, w * hv.x);
    atomicAdd(o + 1, w * hv.y);
    atomicAdd(o + 2, w * hv.z);
    atomicAdd(o + 3, w * hv.w);
}

// ------------------------------------------------- head: logits = [x1,x2] @ Wl^T + bl -> log_softmax
// One wave per node row; 128-dim concat row staged in LDS; 40 output channels:
// lanes 0-31 compute c=lane, lanes 0-7 additionally c=32+lane; wave-shuffle reductions.
__global__ __launch_bounds__(256)
void head_kernel(const float* __restrict__ x1, const float* __restrict__ x2,
                 const float* __restrict__ Wl, const float* __restrict__ bl,
                 float* __restrict__ out) {
    __shared__ float xs[8][128];
    int lane = threadIdx.x & 31;
    int wave = threadIdx.x >> 5;
    int row  = blockIdx.x * 8 + wave;     // 12500 blocks exact

    int f = lane * 4;                     // 0..124
    const float* src = (f < 64) ? (x1 + (size_t)row * HIDC + f)
                                : (x2 + (size_t)row * HIDC + (f - 64));
    *(float4*)&xs[wave][f] = *(const float4*)src;
    __syncthreads();

    float acc1 = bl[lane];
    {
        const float* wrow = Wl + (size_t)lane * 128;
#pragma unroll 8
        for (int k = 0; k < 128; ++k) acc1 += xs[wave][k] * wrow[k];
    }
    float acc2 = 0.0f;
    if (lane < 8) {
        acc2 = bl[32 + lane];
        const float* wrow = Wl + (size_t)(32 + lane) * 128;
#pragma unroll 8
        for (int k = 0; k < 128; ++k) acc2 += xs[wave][k] * wrow[k];
    }

    float m = acc1;
    if (lane < 8) m = fmaxf(m, acc2);
    for (int off = 16; off >= 1; off >>= 1) m = fmaxf(m, __shfl_xor(m, off, 32));

    float s = __expf(acc1 - m) + ((lane < 8) ? __expf(acc2 - m) : 0.0f);
    for (int off = 16; off >= 1; off >>= 1) s += __shfl_xor(s, off, 32);

    float lse = m + __logf(s);
    out[(size_t)row * OUTC + lane] = acc1 - lse;
    if (lane < 8) out[(size_t)row * OUTC + 32 + lane] = acc2 - lse;
}

// ---------------------------------------------------------------- launcher
extern "C" void kernel_launch(void* const* d_in, const int* in_sizes, int n_in,
                              void* d_out, int out_size, void* d_ws, size_t ws_size,
                              hipStream_t stream) {
    (void)in_sizes; (void)n_in; (void)out_size; (void)ws_size;
    const float* x0   = (const float*)d_in[0];
    const int*   eidx = (const int*)  d_in[1];   // [2, E]: rows then cols
    const float* W1   = (const float*)d_in[2];
    const float* b1   = (const float*)d_in[3];
    const float* Wr1  = (const float*)d_in[4];
    const float* br1  = (const float*)d_in[5];
    const float* W2   = (const float*)d_in[6];
    const float* b2   = (const float*)d_in[7];
    const float* Wr2  = (const float*)d_in[8];
    const float* br2  = (const float*)d_in[9];
    const float* Wl   = (const float*)d_in[10];
    const float* bl   = (const float*)d_in[11];
    float* out = (float*)d_out;

    // workspace layout (floats)
    float* ws  = (float*)d_ws;
    const size_t NPAD = 100352;                  // N rounded up for alignment
    float* dis = ws;                             // deg -> d^-1/2 in place
    float* h   = ws + NPAD;                      // N x 64
    float* x1  = h  + (size_t)N_NODES * HIDC;    // N x 64
    float* x2  = x1 + (size_t)N_NODES * HIDC;    // N x 64

    const int elemBlocks = (N_NODES * HIDC + 255) / 256;   // 25000
    const int gemmBlocks = (N_NODES / 16) * 4 / 8;         // 3125
    const int scatBlocks = (N_EDGES * 16) / 256;           // 100000

    // normalization weights
    zero_f32     <<<(N_NODES + 255) / 256, 256, 0, stream>>>(dis, N_NODES);
    degree_kernel<<<(N_EDGES + 255) / 256, 256, 0, stream>>>(eidx, dis, N_EDGES);
    rsqrt_kernel <<<(N_NODES + 255) / 256, 256, 0, stream>>>(dis, N_NODES);

    // layer 1
    gemm64_wmma  <<<gemmBlocks, 256, 0, stream>>>(x0, W1,  b1,  h);
    gemm64_wmma  <<<gemmBlocks, 256, 0, stream>>>(x0, Wr1, br1, x1);
    scatter_kernel<<<scatBlocks, 256, 0, stream>>>(eidx, dis, h, x1);
    relu_kernel  <<<elemBlocks, 256, 0, stream>>>(x1, N_NODES * HIDC);

    // layer 2
    gemm64_wmma  <<<gemmBlocks, 256, 0, stream>>>(x1, W2,  b2,  h);
    gemm64_wmma  <<<gemmBlocks, 256, 0, stream>>>(x1, Wr2, br2, x2);
    scatter_kernel<<<scatBlocks, 256, 0, stream>>>(eidx, dis, h, x2);
    relu_kernel  <<<elemBlocks, 256, 0, stream>>>(x2, N_NODES * HIDC);

    // head + log_softmax
    head_kernel  <<<N_NODES / 8, 256, 0, stream>>>(x1, x2, Wl, bl, out);
}